// Decoder_35321811042426
// MI455X (gfx1250) — compile-verified
//
#include <hip/hip_runtime.h>
#include <cmath>

// ---------------------------------------------------------------------------
// Autoregressive LSTM decoder, H=2048, T=out_size/H (512).
// gates_t = (W_ih + W_hh) @ h_{t-1} + (b_ih + b_hh)   for t >= 1  (x_t == h_{t-1})
// gates_0 = W_ih @ x0 + (b_ih + b_hh)                 (h0 = c0 = 0)
//
// Bound by weight streaming: 67 MB (W_sum fp32) per step, resident in 192 MB L2.
// Persistent kernel, 1 device-wide barrier per step. Core reduction runs on
// v_wmma_f32_16x16x4_f32 with W_sum pre-swizzled into A-matrix register order
// so all global loads are fully coalesced b128s. Dual accumulators + unroll 8
// keep ~4 KB of loads in flight per wave and break the WMMA C-chain.
// ---------------------------------------------------------------------------

constexpr int kH   = 2048;
constexpr int kNWG = 128;   // 2048 / 16 h-elements per WG
constexpr int kTPB = 256;   // 8 wave32s

typedef float v2f __attribute__((ext_vector_type(2)));
typedef float v4f __attribute__((ext_vector_type(4)));
typedef float v8f __attribute__((ext_vector_type(8)));

__device__ __forceinline__ float sigf(float x) { return 1.0f / (1.0f + expf(-x)); }

// ---------------------------------------------------------------------------
// Pre-swizzle W_sum = W_ih + W_hh into V_WMMA_F32_16X16X4_F32 A-matrix order.
// Block (mt, ks) = 16 rows x 8 cols (two K=4 tiles), 128 floats.
// Lane l owns 4 consecutive floats: {tile0:(K d,d+1), tile1:(K d,d+1)} of row
// (mt*16 + l%16), where d = (l/16)*2  (ISA 7.12.2, 32-bit A 16x4 layout).
// Hot-loop load = global_load_b128 at (blk*128 + lane*4): perfectly coalesced.
// ---------------------------------------------------------------------------
__global__ void swizzle_wsum(const float* __restrict__ Wih,
                             const float* __restrict__ Whh,
                             float* __restrict__ SW) {
  size_t q   = (size_t)blockIdx.x * blockDim.x + threadIdx.x; // < 512*256*32
  int   lane = (int)(q & 31);
  size_t blk = q >> 5;               // mt*256 + ks
  int   ks   = (int)(blk & 255);
  int   mt   = (int)(blk >> 8);      // 0..511
  int   m    = mt * 16 + (lane & 15);
  int   d    = (lane >> 4) * 2;
  int   cb   = ks * 8;
  const float* a = Wih + (size_t)m * kH;
  const float* b = Whh + (size_t)m * kH;
  v4f o;
  o.x = a[cb + d]         + b[cb + d];
  o.y = a[cb + d + 1]     + b[cb + d + 1];
  o.z = a[cb + 4 + d]     + b[cb + 4 + d];
  o.w = a[cb + 4 + d + 1] + b[cb + 4 + d + 1];
  *(v4f*)(SW + blk * 128 + (size_t)lane * 4) = o;
}

__global__ void __launch_bounds__(kTPB, 1)
lstm_decoder(const float* __restrict__ X,
             const float* __restrict__ Wih,
             const float* __restrict__ Whh,
             const float* __restrict__ bih,
             const float* __restrict__ bhh,
             float* __restrict__ out,          // T x H; also the h history
             const float* __restrict__ SW,     // swizzled W_sum, or nullptr
             unsigned* __restrict__ bar_cnt,
             unsigned* __restrict__ bar_gen,
             int T) {
  __shared__ __align__(16) float h_lds[kH];
  __shared__ float p_lds[4][2][16];   // [gate][k-half][j-local] partial sums
  __shared__ float c_lds[16];         // persistent cell state for this WG

  const int tid  = threadIdx.x;
  const int wg   = blockIdx.x;
  const int lane = tid & 31;
  const int wv   = tid >> 5;          // wave 0..7

  if (tid < 16) c_lds[tid] = 0.0f;

  for (int t = 0; t < T; ++t) {
    // ---- 1) stage this step's input vector (x0 or h_{t-1}) into LDS -------
    const float* src = (t == 0) ? X : (out + (size_t)(t - 1) * kH);
    for (int i = tid; i < kH; i += kTPB)
      h_lds[i] = __hip_atomic_load(src + i, __ATOMIC_RELAXED, __HIP_MEMORY_SCOPE_AGENT);
    __syncthreads();

    // ---- 2) 64 gate rows for this WG ---------------------------------------
    if (t > 0 && SW) {
      // WMMA path: wave = (gate g, K-half kh); accumulate 128 K-tile pairs.
      const int g  = wv >> 1;
      const int kh = wv & 1;
      const int mt = g * 128 + wg;
      const float* swp = SW + (((size_t)mt * 256 + (size_t)kh * 128) * 128) + (size_t)lane * 4;
      const int    d   = (lane >> 4) * 2;
      const float* hb  = h_lds + kh * 1024 + d;   // B operand: broadcast ds_load_b64
      v8f acc0 = {0.f, 0.f, 0.f, 0.f, 0.f, 0.f, 0.f, 0.f};
      v8f acc1 = {0.f, 0.f, 0.f, 0.f, 0.f, 0.f, 0.f, 0.f};
#pragma unroll 8
      for (int ks = 0; ks < 128; ++ks) {
        v4f a  = *(const v4f*)swp;  swp += 128;   // coalesced b128: two A tiles
        v2f b0 = *(const v2f*)(hb);               // h[cb+d .. cb+d+1]
        v2f b1 = *(const v2f*)(hb + 4);           // h[cb+4+d .. cb+5+d]
        hb += 8;
        v2f a0 = { a.x, a.y };
        v2f a1 = { a.z, a.w };
        acc0 = __builtin_amdgcn_wmma_f32_16x16x4_f32(false, a0, false, b0,
                                                     (short)0, acc0, false, false);
        acc1 = __builtin_amdgcn_wmma_f32_16x16x4_f32(false, a1, false, b1,
                                                     (short)0, acc1, false, false);
      }
      // D col 0: lane 0 holds M=r in VGPR r, lane 16 holds M=r+8.
      if (lane == 0 || lane == 16) {
        int off = (lane >> 4) * 8;
#pragma unroll
        for (int r = 0; r < 8; ++r) p_lds[g][kh][off + r] = acc0[r] + acc1[r];
      }
    } else {
      // VALU path: step 0 (W_ih only) and no-workspace fallback (W_ih+W_hh).
      const bool both = (t > 0);
#pragma unroll
      for (int rr = 0; rr < 8; ++rr) {
        int lr   = wv * 8 + rr;                 // local gate-row 0..63
        int gate = lr >> 4;
        int jl   = lr & 15;
        size_t grow = (size_t)gate * kH + (size_t)wg * 16 + jl;
        const float* wa = Wih + grow * kH + (size_t)lane * 4;
        const float* wb = Whh + grow * kH + (size_t)lane * 4;
        const float* hl = h_lds + lane * 4;
        float acc = 0.f;
#pragma unroll
        for (int k = 0; k < 16; ++k) {
          v4f w = *(const v4f*)(wa + k * 128);
          if (both) {
            v4f w2 = *(const v4f*)(wb + k * 128);
            w.x += w2.x; w.y += w2.y; w.z += w2.z; w.w += w2.w;
          }
          v4f hv = *(const v4f*)(hl + k * 128);
          acc += w.x * hv.x + w.y * hv.y + w.z * hv.z + w.w * hv.w;
        }
#pragma unroll
        for (int m = 16; m >= 1; m >>= 1) acc += __shfl_xor(acc, m, 32);
        if (lane == 0) { p_lds[gate][0][jl] = acc; p_lds[gate][1][jl] = 0.f; }
      }
    }
    __syncthreads();

    // ---- 3) elementwise LSTM cell update (16 h-elements, wave 0) -----------
    if (tid < 16) {
      int j = wg * 16 + tid;
      float gi = p_lds[0][0][tid] + p_lds[0][1][tid] + bih[j]          + bhh[j];
      float gf = p_lds[1][0][tid] + p_lds[1][1][tid] + bih[kH + j]     + bhh[kH + j];
      float gg = p_lds[2][0][tid] + p_lds[2][1][tid] + bih[2 * kH + j] + bhh[2 * kH + j];
      float go = p_lds[3][0][tid] + p_lds[3][1][tid] + bih[3 * kH + j] + bhh[3 * kH + j];
      float c  = sigf(gf) * c_lds[tid] + sigf(gi) * tanhf(gg);
      float h  = sigf(go) * tanhf(c);
      c_lds[tid] = c;
      __hip_atomic_store(out + (size_t)t * kH + j, h,
                         __ATOMIC_RELAXED, __HIP_MEMORY_SCOPE_AGENT);
    }

    // ---- 4) device-wide barrier (sense-reversing, L2 atomics) --------------
    __threadfence();
    __syncthreads();
    if (tid == 0) {
      unsigned g0 = __hip_atomic_load(bar_gen, __ATOMIC_RELAXED, __HIP_MEMORY_SCOPE_AGENT);
      unsigned a  = __hip_atomic_fetch_add(bar_cnt, 1u,
                                           __ATOMIC_ACQ_REL, __HIP_MEMORY_SCOPE_AGENT);
      if (a == (unsigned)(kNWG - 1)) {
        __hip_atomic_store(bar_cnt, 0u, __ATOMIC_RELAXED, __HIP_MEMORY_SCOPE_AGENT);
        __hip_atomic_fetch_add(bar_gen, 1u, __ATOMIC_ACQ_REL, __HIP_MEMORY_SCOPE_AGENT);
      } else {
        while (__hip_atomic_load(bar_gen, __ATOMIC_ACQUIRE, __HIP_MEMORY_SCOPE_AGENT) == g0)
          __builtin_amdgcn_s_sleep(1);
      }
    }
    __syncthreads();
  }
}

extern "C" void kernel_launch(void* const* d_in, const int* in_sizes, int n_in,
                              void* d_out, int out_size, void* d_ws, size_t ws_size,
                              hipStream_t stream) {
  const float* X   = (const float*)d_in[0];
  const float* Wih = (const float*)d_in[1];
  const float* Whh = (const float*)d_in[2];
  const float* bih = (const float*)d_in[3];
  const float* bhh = (const float*)d_in[4];
  float* out = (float*)d_out;
  const int T = out_size / kH;            // sentence_length (device-scalar avoided)

  unsigned char* ws = (unsigned char*)d_ws;
  unsigned* bar_cnt = (unsigned*)ws;
  unsigned* bar_gen = (unsigned*)(ws + 128);
  const size_t SW_OFF   = 512;
  const size_t SW_BYTES = (size_t)512 * 256 * 128 * sizeof(float); // 64 MiB
  float* SW = (ws_size >= SW_OFF + SW_BYTES) ? (float*)(ws + SW_OFF) : nullptr;

  hipMemsetAsync(d_ws, 0, 256, stream);   // barrier state (graph-capturable)
  if (SW) {
    // 512*256 blocks * 32 lanes, 4 floats per thread
    swizzle_wsum<<<16384, 256, 0, stream>>>(Wih, Whh, SW);
  }
  lstm_decoder<<<kNWG, kTPB, 0, stream>>>(X, Wih, Whh, bih, bhh, out,
                                          SW, bar_cnt, bar_gen, T);
}